// HGT_67834713473480
// MI455X (gfx1250) — compile-verified
//
#include <hip/hip_runtime.h>
#include <math.h>
#include <stdint.h>

#define HEADS 8
#define DKV 16
#define DMODEL 128

typedef float v2f __attribute__((ext_vector_type(2)));
typedef float v8f __attribute__((ext_vector_type(8)));

// ---------------------------------------------------------------------------
// CDNA5 WMMA helper: D = A(16x4 f32) * B(4x16 f32) + C(16x16 f32)
// A frag: lane m=lane&15, k0=(lane>>4)*2 ; a.x=A[m][k0], a.y=A[m][k0+1]
// B frag: lane n=lane&15, k0=(lane>>4)*2 ; b.x=B[k0][n], b.y=B[k0+1][n]
// C frag: VGPR v, lane l: row = v + 8*(l>=16), col = l&15
// ---------------------------------------------------------------------------
__device__ inline v8f wmma4(v2f a, v2f b, v8f c) {
  return __builtin_amdgcn_wmma_f32_16x16x4_f32(false, a, false, b, (short)0, c,
                                               false, false);
}

// Async global->LDS copy (gfx1250): 16B per lane, tracked by ASYNCcnt.
// lds_off is the LDS byte address (low 32 bits of the generic shared pointer,
// per ISA flat->LDS truncation).
__device__ inline void async_g2lds_b128(uint32_t lds_off, const float* gaddr) {
  asm volatile("global_load_async_to_lds_b128 %0, %1, off"
               :
               : "v"(lds_off), "v"(gaddr)
               : "memory");
}
__device__ inline void wait_asynccnt0() {
  asm volatile("s_wait_asynccnt 0x0" ::: "memory");
}

// ---------------------------------------------------------------------------
// Build combined K/V weights: Wc[mat] = Wk_or_Wv[st(r)] @ blockdiag(rel[r])
// mat 0..3 -> K with rel_att, mat 4..7 -> V with rel_msg. row==128 -> bias.
// ---------------------------------------------------------------------------
__global__ void combine_weights(const float* __restrict__ Wk,
                                const float* __restrict__ bk,
                                const float* __restrict__ Wv,
                                const float* __restrict__ bv,
                                const float* __restrict__ rel_att,
                                const float* __restrict__ rel_msg,
                                float* __restrict__ Wc,
                                float* __restrict__ bc) {
  const int mat = blockIdx.y;   // 0..7
  const int row = blockIdx.x;   // 0..128 (128 == bias row)
  const int j = threadIdx.x;    // 0..127
  const int r = mat & 3;
  const bool isv = mat >= 4;
  const int st = r & 1;         // REL src types: 0,1,0,1
  const float* W = isv ? Wv : Wk;
  const float* b = isv ? bv : bk;
  const float* Rm = isv ? rel_msg : rel_att;
  const int h = j >> 4, jj = j & 15;
  const float* Rblk = Rm + (size_t)((r * HEADS + h) * DKV) * DKV + jj;
  float s = 0.f;
  if (row < DMODEL) {
    const float* wrow =
        W + (size_t)st * DMODEL * DMODEL + (size_t)row * DMODEL + h * DKV;
#pragma unroll
    for (int d = 0; d < DKV; ++d) s += wrow[d] * Rblk[(size_t)d * DKV];
    Wc[(size_t)mat * DMODEL * DMODEL + (size_t)row * DMODEL + j] = s;
  } else {
    const float* brow = b + (size_t)st * DMODEL + h * DKV;
#pragma unroll
    for (int d = 0; d < DKV; ++d) s += brow[d] * Rblk[(size_t)d * DKV];
    bc[(size_t)mat * DMODEL + j] = s;
  }
}

__global__ void fill_f32(float* __restrict__ p, float v, long n) {
  long i = (long)blockIdx.x * blockDim.x + threadIdx.x;
  if (i < n) p[i] = v;
}

// ---------------------------------------------------------------------------
// C[N x 128] = A[N x 128] @ W[128 x 128] + bias  (fp32 WMMA 16x16x4)
// 8 waves per block; wave w owns the 16x16 C tile at columns 16w.
// A tile (16x128, contiguous 8KB) staged to LDS once via async-to-LDS.
// ---------------------------------------------------------------------------
__global__ __launch_bounds__(256) void gemm_bias_128(
    const float* __restrict__ A, const float* __restrict__ W,
    const float* __restrict__ bias, float* __restrict__ C, int nrows) {
  __shared__ float atile[16 * DMODEL];  // 8 KB
  const int tid = threadIdx.x;
  const int wave = tid >> 5;
  const int lane = tid & 31;
  const int rowbase = blockIdx.x << 4;
  if (rowbase >= nrows) return;

  {  // cooperative async copy: 512 x b128, 2 per thread
    const float* gsrc = A + (size_t)rowbase * DMODEL;
    const uint32_t lbase = (uint32_t)(uintptr_t)atile;
    async_g2lds_b128(lbase + tid * 16u, gsrc + tid * 4);
    async_g2lds_b128(lbase + (tid + 256) * 16u, gsrc + (tid + 256) * 4);
    wait_asynccnt0();
  }
  __syncthreads();

  const int colbase = wave << 4;
  const int m = lane & 15;
  const int k0 = (lane >> 4) << 1;  // 0 or 2
  v8f acc = {};
  const float* arow = atile + m * DMODEL + k0;  // LDS reads (ds_load_b64)
  const float* wcol = W + (size_t)k0 * DMODEL + colbase + m;
  for (int k = 0; k < DMODEL; k += 4) {
    v2f a;
    a.x = arow[k];
    a.y = arow[k + 1];
    v2f b;
    b.x = wcol[(size_t)k * DMODEL];
    b.y = wcol[(size_t)(k + 1) * DMODEL];
    acc = wmma4(a, b, acc);
  }
  const float bval = bias[colbase + m];
  const int rowoff = (lane >> 4) << 3;  // 0 or 8
#pragma unroll
  for (int v = 0; v < 8; ++v)
    C[(size_t)(rowbase + rowoff + v) * DMODEL + colbase + m] = acc[v] + bval;
}

// ---------------------------------------------------------------------------
// Edge softmax pass 1: score[e][h] = <q[dst],k[src]>_h * pri[h] / 4,
// atomic segment-max into m[dst][h]. One wave32 per edge.
// ---------------------------------------------------------------------------
__device__ inline float atomicMaxF(float* addr, float v) {
  if (v >= 0.f)
    return __int_as_float(atomicMax((int*)addr, __float_as_int(v)));
  return __uint_as_float(atomicMin((unsigned int*)addr, __float_as_uint(v)));
}

__global__ __launch_bounds__(256) void edge_score(
    const float* __restrict__ q, const float* __restrict__ k,
    const int* __restrict__ src, const int* __restrict__ dst,
    const float* __restrict__ pri, float* __restrict__ score,
    float* __restrict__ m, int E) {
  const int wave = threadIdx.x >> 5, lane = threadIdx.x & 31;
  const int e = blockIdx.x * 8 + wave;
  if (e >= E) return;
  const int s = src[e], d = dst[e];
  const float4 qv = *(const float4*)(q + (size_t)d * DMODEL + lane * 4);
  const float4 kv = *(const float4*)(k + (size_t)s * DMODEL + lane * 4);
  float p = qv.x * kv.x + qv.y * kv.y + qv.z * kv.z + qv.w * kv.w;
  p += __shfl_xor(p, 1, 32);
  p += __shfl_xor(p, 2, 32);
  const int h = lane >> 2;
  if ((lane & 3) == 0) {
    const float sc = p * pri[h] * 0.25f;  // 1/sqrt(DK)
    score[(size_t)e * HEADS + h] = sc;
    atomicMaxF(m + (size_t)d * HEADS + h, sc);
  }
}

// Pass 2: sc <- exp(sc - m[dst]); den[dst] += sc   (in place over score buf)
__global__ void edge_exp(float* __restrict__ sc, const float* __restrict__ m,
                         float* __restrict__ den, const int* __restrict__ dst,
                         int E) {
  const int i = blockIdx.x * blockDim.x + threadIdx.x;
  if (i >= E * HEADS) return;
  const int e = i >> 3, h = i & 7;
  const int d = dst[e];
  const float ex = __expf(sc[i] - m[(size_t)d * HEADS + h]);
  sc[i] = ex;
  atomicAdd(den + (size_t)d * HEADS + h, ex);
}

// Pass 3: agg[dst] += v[src] * (ex / den[dst]). One wave32 per edge.
__global__ __launch_bounds__(256) void edge_aggregate(
    const float* __restrict__ vproj, const float* __restrict__ ex,
    const float* __restrict__ den, const int* __restrict__ src,
    const int* __restrict__ dst, float* __restrict__ agg, int E) {
  const int wave = threadIdx.x >> 5, lane = threadIdx.x & 31;
  const int e = blockIdx.x * 8 + wave;
  if (e >= E) return;
  const int s = src[e], d = dst[e];
  const int h = lane >> 2;
  const float attn = ex[(size_t)e * HEADS + h] / den[(size_t)d * HEADS + h];
  const float4 vv = *(const float4*)(vproj + (size_t)s * DMODEL + lane * 4);
  float* ap = agg + (size_t)d * DMODEL + lane * 4;
  atomicAdd(ap + 0, vv.x * attn);
  atomicAdd(ap + 1, vv.y * attn);
  atomicAdd(ap + 2, vv.z * attn);
  atomicAdd(ap + 3, vv.w * attn);
}

// ---------------------------------------------------------------------------
// Output: trans = (agg/2) @ Wa + ba ; out = trans*a + h*(1-a) ; LayerNorm.
// agg tile staged via async-to-LDS; LN stats via LDS tile + lane shuffles.
// ---------------------------------------------------------------------------
__global__ __launch_bounds__(256) void out_transform(
    const float* __restrict__ agg, const float* __restrict__ hin,
    const float* __restrict__ Wa, const float* __restrict__ ba,
    const float* __restrict__ g, const float* __restrict__ bb,
    const float* __restrict__ skip, float* __restrict__ out, int nrows) {
  __shared__ float atile[16 * DMODEL];  // staged agg tile
  __shared__ float tile[16 * DMODEL];   // skip-blended rows for LN
  const int tid = threadIdx.x;
  const int wave = tid >> 5, lane = tid & 31;
  const int rowbase = blockIdx.x << 4;
  if (rowbase >= nrows) return;

  {  // cooperative async copy of agg tile
    const float* gsrc = agg + (size_t)rowbase * DMODEL;
    const uint32_t lbase = (uint32_t)(uintptr_t)atile;
    async_g2lds_b128(lbase + tid * 16u, gsrc + tid * 4);
    async_g2lds_b128(lbase + (tid + 256) * 16u, gsrc + (tid + 256) * 4);
    wait_asynccnt0();
  }
  __syncthreads();

  const int colbase = wave << 4;
  const int m = lane & 15;
  const int k0 = (lane >> 4) << 1;
  const float alpha = 1.f / (1.f + __expf(-skip[0]));
  v8f acc = {};
  const float* arow = atile + m * DMODEL + k0;
  const float* wcol = Wa + (size_t)k0 * DMODEL + colbase + m;
  for (int k = 0; k < DMODEL; k += 4) {
    v2f a;
    a.x = arow[k] * 0.5f;  // mean over the 2 incoming relations
    a.y = arow[k + 1] * 0.5f;
    v2f b;
    b.x = wcol[(size_t)k * DMODEL];
    b.y = wcol[(size_t)(k + 1) * DMODEL];
    acc = wmma4(a, b, acc);
  }
  const float bias = ba[colbase + m];
  const int rowoff = (lane >> 4) << 3;
#pragma unroll
  for (int v = 0; v < 8; ++v) {
    const int rl = rowoff + v;
    const float t = acc[v] + bias;
    const float hv = hin[(size_t)(rowbase + rl) * DMODEL + colbase + m];
    tile[rl * DMODEL + colbase + m] = t * alpha + hv * (1.f - alpha);
  }
  __syncthreads();
  // LayerNorm: wave w handles rows 2w and 2w+1 (128 cols = 32 lanes x float4)
  for (int rr = wave * 2; rr < wave * 2 + 2; ++rr) {
    const float4 x = *(const float4*)(tile + rr * DMODEL + lane * 4);
    float sum = x.x + x.y + x.z + x.w;
    float sq = x.x * x.x + x.y * x.y + x.z * x.z + x.w * x.w;
    for (int o = 1; o < 32; o <<= 1) {
      sum += __shfl_xor(sum, o, 32);
      sq += __shfl_xor(sq, o, 32);
    }
    const float mu = sum * (1.f / DMODEL);
    const float var = sq * (1.f / DMODEL) - mu * mu;
    const float rstd = rsqrtf(var + 1e-5f);
    const float4 gg = *(const float4*)(g + lane * 4);
    const float4 b4 = *(const float4*)(bb + lane * 4);
    float4 o4;
    o4.x = (x.x - mu) * rstd * gg.x + b4.x;
    o4.y = (x.y - mu) * rstd * gg.y + b4.y;
    o4.z = (x.z - mu) * rstd * gg.z + b4.z;
    o4.w = (x.w - mu) * rstd * gg.w + b4.w;
    *(float4*)(out + (size_t)(rowbase + rr) * DMODEL + lane * 4) = o4;
  }
}

// ---------------------------------------------------------------------------
extern "C" void kernel_launch(void* const* d_in, const int* in_sizes, int n_in,
                              void* d_out, int out_size, void* d_ws,
                              size_t ws_size, hipStream_t stream) {
  const float* h0 = (const float*)d_in[0];
  const float* h1 = (const float*)d_in[1];
  const int* srcs[4] = {(const int*)d_in[2], (const int*)d_in[4],
                        (const int*)d_in[6], (const int*)d_in[8]};
  const int* dsts[4] = {(const int*)d_in[3], (const int*)d_in[5],
                        (const int*)d_in[7], (const int*)d_in[9]};
  const float* Wk = (const float*)d_in[10];
  const float* bk = (const float*)d_in[11];
  const float* Wq = (const float*)d_in[12];
  const float* bq = (const float*)d_in[13];
  const float* Wv = (const float*)d_in[14];
  const float* bv = (const float*)d_in[15];
  const float* Wa = (const float*)d_in[16];
  const float* ba = (const float*)d_in[17];
  const float* rel_att = (const float*)d_in[18];
  const float* rel_msg = (const float*)d_in[19];
  const float* rel_pri = (const float*)d_in[20];
  const float* skip = (const float*)d_in[21];
  const float* ln_g = (const float*)d_in[22];
  const float* ln_b = (const float*)d_in[23];

  const int N = in_sizes[0] / DMODEL;
  const int E = in_sizes[2];
  const float* hs[2] = {h0, h1};
  const int st_of[4] = {0, 1, 0, 1};
  const int dt_of[4] = {1, 0, 0, 1};

  // ---- workspace carving (floats) ----
  float* ws = (float*)d_ws;
  const size_t nd = (size_t)N * DMODEL;
  const size_t nh = (size_t)N * HEADS;
  const size_t eh = (size_t)E * HEADS;
  float* q[2];
  q[0] = ws; q[1] = ws + nd; ws += 2 * nd;
  float* krel[4];
  for (int r = 0; r < 4; ++r) { krel[r] = ws; ws += nd; }
  float* vrel[4];
  for (int r = 0; r < 4; ++r) { vrel[r] = ws; ws += nd; }
  float* score[4];
  for (int r = 0; r < 4; ++r) { score[r] = ws; ws += eh; }
  float* mbuf = ws; ws += 4 * nh;       // 4 relations, contiguous
  float* den = ws; ws += 4 * nh;        // 4 relations, contiguous
  float* aggb = ws; ws += 2 * nd;       // 2 dst types, contiguous
  float* Wc = ws; ws += 8 * DMODEL * DMODEL;
  float* bc = ws; ws += 8 * DMODEL;

  // ---- stage 0: combined weights + accumulator init ----
  combine_weights<<<dim3(129, 8), 128, 0, stream>>>(Wk, bk, Wv, bv, rel_att,
                                                    rel_msg, Wc, bc);
  {
    long n1 = (long)(4 * nh);
    fill_f32<<<(n1 + 255) / 256, 256, 0, stream>>>(mbuf, -INFINITY, n1);
    fill_f32<<<(n1 + 255) / 256, 256, 0, stream>>>(den, 0.f, n1);
    long n2 = (long)(2 * nd);
    fill_f32<<<(n2 + 255) / 256, 256, 0, stream>>>(aggb, 0.f, n2);
  }

  // ---- stage 1: projections (fp32 WMMA GEMMs, async-LDS staged A) ----
  const int gtiles = (N + 15) / 16;  // N=50000 -> 3125 exact
  gemm_bias_128<<<gtiles, 256, 0, stream>>>(h0, Wq, bq, q[0], N);
  gemm_bias_128<<<gtiles, 256, 0, stream>>>(h1, Wq + DMODEL * DMODEL,
                                            bq + DMODEL, q[1], N);
  for (int r = 0; r < 4; ++r)
    gemm_bias_128<<<gtiles, 256, 0, stream>>>(
        hs[st_of[r]], Wc + (size_t)r * DMODEL * DMODEL, bc + r * DMODEL,
        krel[r], N);
  for (int r = 0; r < 4; ++r)
    gemm_bias_128<<<gtiles, 256, 0, stream>>>(
        hs[st_of[r]], Wc + (size_t)(4 + r) * DMODEL * DMODEL,
        bc + (4 + r) * DMODEL, vrel[r], N);

  // ---- stage 2: edge softmax + aggregation ----
  const int eb = (E + 7) / 8;
  for (int r = 0; r < 4; ++r)
    edge_score<<<eb, 256, 0, stream>>>(q[dt_of[r]], krel[r], srcs[r], dsts[r],
                                       rel_pri + r * HEADS, score[r],
                                       mbuf + r * nh, E);
  for (int r = 0; r < 4; ++r)
    edge_exp<<<((long)eh + 255) / 256, 256, 0, stream>>>(
        score[r], mbuf + r * nh, den + r * nh, dsts[r], E);
  for (int r = 0; r < 4; ++r)
    edge_aggregate<<<eb, 256, 0, stream>>>(vrel[r], score[r], den + r * nh,
                                           srcs[r], dsts[r],
                                           aggb + (size_t)dt_of[r] * nd, E);

  // ---- stage 3: output projection + skip + LayerNorm ----
  float* outp = (float*)d_out;
  for (int t = 0; t < 2; ++t)
    out_transform<<<gtiles, 256, 0, stream>>>(
        aggb + (size_t)t * nd, hs[t], Wa + (size_t)t * DMODEL * DMODEL,
        ba + t * DMODEL, ln_g + t * DMODEL, ln_b + t * DMODEL, skip + t,
        outp + (size_t)t * nd, N);
}